// Decoder_21732534518004
// MI455X (gfx1250) — compile-verified
//
#include <hip/hip_runtime.h>
#include <hip/hip_bf16.h>
#include <math.h>

// Problem constants (from reference setup_inputs)
#define BB 256      // batch
#define NN 1024     // nodes
#define EE 256      // embed
#define HH 8        // heads
#define DH 32       // head dim
#define TT 64       // decode steps
#define MM (BB*NN)  // GEMM rows = 262144
#define CLIPC 10.0f

typedef float v2f __attribute__((ext_vector_type(2)));
typedef float v8f __attribute__((ext_vector_type(8)));

// ---------------------------------------------------------------------------
// Precompute GEMM:  out[M, 256] = X[M,256] @ W[:, colOff:colOff+256] + bias
// Block: 128 threads (4 wave32s). Block tile 64(M) x 64(N), K-chunks of 64.
// Each wave owns a 16(M) x 64(N) strip = 4 accumulators of 16x16 f32.
// Uses V_WMMA_F32_16X16X4_F32 (exact f32 — greedy argmax demands f32 parity).
// ---------------------------------------------------------------------------
__global__ __launch_bounds__(128)
void gemm_wmma_f32(const float* __restrict__ X, const float* __restrict__ W,
                   int ldw, int colOff, const float* __restrict__ bias,
                   int biasOff, float* __restrict__ out, int M, int K, int Nw)
{
    __shared__ float Xs[64][68];   // +4 pad (272B row stride, 16B aligned)
    __shared__ float Ws[64][68];

    const int tid  = threadIdx.x;
    const int wave = tid >> 5;
    const int lane = tid & 31;
    const long blockM = (long)blockIdx.x * 64;
    const int  blockN = blockIdx.y * 64;

    v8f acc[4];
#pragma unroll
    for (int s = 0; s < 4; ++s) acc[s] = (v8f){0,0,0,0,0,0,0,0};

    const int ml = lane & 15;           // M (for A) / N (for B) within 16
    const int kk = (lane < 16) ? 0 : 2; // K sub-offset per ISA layout

    for (int k0 = 0; k0 < K; k0 += 64) {
        // Cooperative load: 64x64 of X and W as float4 (1024 vec4 each)
        for (int f = tid; f < 1024; f += 128) {
            const int row = f >> 4;
            const int cg  = (f & 15) << 2;
            float4 vx = *(const float4*)(X + (blockM + row) * (long)K + (k0 + cg));
            *(float4*)&Xs[row][cg] = vx;
            float4 vw = *(const float4*)(W + (long)(k0 + row) * ldw + (colOff + blockN + cg));
            *(float4*)&Ws[row][cg] = vw;
        }
        __syncthreads();

#pragma unroll
        for (int k = 0; k < 64; k += 4) {
            // A fragment: 16x4 f32 (lanes 0-15: K=k+0/1, lanes 16-31: K=k+2/3)
            v2f a;
            a.x = Xs[wave * 16 + ml][k + kk];
            a.y = Xs[wave * 16 + ml][k + kk + 1];
#pragma unroll
            for (int s = 0; s < 4; ++s) {
                // B fragment: 4x16 f32, mirrored layout
                v2f bf;
                bf.x = Ws[k + kk][s * 16 + ml];
                bf.y = Ws[k + kk + 1][s * 16 + ml];
#if __has_builtin(__builtin_amdgcn_wmma_f32_16x16x4_f32)
                acc[s] = __builtin_amdgcn_wmma_f32_16x16x4_f32(
                    false, a, false, bf, (short)0, acc[s], false, false);
#else
                // fallback (should not trigger on gfx1250): scalar FMA rank-4 update
                for (int r = 0; r < 8; ++r)
                    acc[s][r] = fmaf(a.x, bf.x, fmaf(a.y, bf.y, acc[s][r]));
#endif
            }
        }
        __syncthreads();
    }

    // D layout: VGPR r -> M = r (+8 for lanes>=16), N = lane&15 within subtile
    const int rbase = (lane < 16) ? 0 : 8;
#pragma unroll
    for (int s = 0; s < 4; ++s) {
        const int col = blockN + s * 16 + ml;
        const float bval = bias[biasOff + col];
#pragma unroll
        for (int r = 0; r < 8; ++r) {
            const long row = blockM + wave * 16 + rbase + r;
            out[row * (long)Nw + col] = acc[s][r] + bval;
        }
    }
}

// ---------------------------------------------------------------------------
// State init: running sum of x over unvisited (all), capacity, visited, count
// ---------------------------------------------------------------------------
__global__ __launch_bounds__(256)
void init_state(const float* __restrict__ x, const float* __restrict__ cap0,
                float* __restrict__ xsum, float* __restrict__ capacity,
                int* __restrict__ current, int* __restrict__ cntArr,
                unsigned char* __restrict__ visited)
{
    const int b = blockIdx.x, tid = threadIdx.x;
    float a = 0.f;
    const float* xb = x + (size_t)b * NN * EE + tid;
    for (int n = 0; n < NN; ++n) a += xb[(size_t)n * EE];   // coalesced
    xsum[(size_t)b * EE + tid] = a;
    for (int n = tid; n < NN; n += 256) visited[(size_t)b * NN + n] = 0;
    if (tid == 0) { capacity[b] = cap0[b]; current[b] = 0; cntArr[b] = NN; }
}

// ---------------------------------------------------------------------------
// One decode step. Grid = B blocks, 256 threads (8 wave32s) per block.
// ---------------------------------------------------------------------------
__global__ __launch_bounds__(256)
void decode_step(const float* __restrict__ x, const float* __restrict__ demand,
                 const float* __restrict__ Wc, const float* __restrict__ bc,
                 const float* __restrict__ Wqkv, const float* __restrict__ bqkv,
                 const float* __restrict__ Wo, const float* __restrict__ bo,
                 const float* __restrict__ Wpq, const float* __restrict__ bpq,
                 const float* __restrict__ kbuf, const float* __restrict__ vbuf,
                 const float* __restrict__ kpbuf,
                 float* __restrict__ xsum, float* __restrict__ capacity,
                 int* __restrict__ current, int* __restrict__ cntArr,
                 unsigned char* __restrict__ visited,
                 float* __restrict__ probs_out, float* __restrict__ route_out,
                 int t)
{
    const int b = blockIdx.x;
    const int tid = threadIdx.x;

    __shared__ float s_mean[EE], s_sel[EE], s_ve[EE], s_q[EE];
    __shared__ float s_cx[EE], s_o[EE], s_qp[EE];
    __shared__ float s_scores[HH][NN];   // 32 KB
    __shared__ float s_u[NN];
    __shared__ unsigned char s_mask[NN];
    __shared__ float s_hsum[HH];
    __shared__ float s_rv[256];
    __shared__ int   s_ri[256];
    __shared__ int   s_allm;

    const float cap = capacity[b];
    const int   cur = current[b];
    const float cnt = (float)cntArr[b];

    // ---- P0: mean embed (incremental sum), selected embed ----
    s_mean[tid] = xsum[(size_t)b * EE + tid] / cnt;
    s_sel[tid]  = x[((size_t)b * NN + cur) * EE + tid];
    if (tid == 0) s_allm = 1;
    __syncthreads();

    // ---- P1: ve = [mean|sel|cap] @ Wc + bc  (thread per output col) ----
    {
        float a = bc[tid];
        for (int i = 0; i < EE; ++i) a = fmaf(s_mean[i], Wc[i * EE + tid], a);
        for (int i = 0; i < EE; ++i) a = fmaf(s_sel[i],  Wc[(EE + i) * EE + tid], a);
        a = fmaf(cap, Wc[2 * EE * EE + tid], a);
        s_ve[tid] = a;
    }
    __syncthreads();

    // ---- P2: q = ve @ Wq_in + bq_in; mask build ----
    {
        float a = bqkv[tid];
        for (int i = 0; i < EE; ++i) a = fmaf(s_ve[i], Wqkv[i * (3 * EE) + tid], a);
        s_q[tid] = a;
    }
    {
        bool anyun = false;
        for (int n = tid; n < NN; n += 256) {
            bool m = (visited[(size_t)b * NN + n] != 0) ||
                     (demand[(size_t)b * NN + n] > cap);
            if (n == 0 && cur == 0) m = true;
            s_mask[n] = m ? 1 : 0;
            if (!m) anyun = true;
        }
        if (anyun) s_allm = 0;
    }
    __syncthreads();
    if (tid == 0 && s_allm) s_mask[0] = 0;   // unmask depot if all masked
    __syncthreads();

    // ---- P3: attention scores, 8 threads per node (one per head) ----
    {
        const int g = tid >> 3, h = tid & 7;
        const float scale = 0.17677669529663687f;   // 1/sqrt(32)
        const float* qh = &s_q[h * DH];
        for (int n = g; n < NN; n += 32) {
            float s;
            if (s_mask[n]) s = -__builtin_inff();
            else {
                const float* kv = kbuf + ((size_t)b * NN + n) * EE + h * DH;
                __builtin_prefetch(kv + 32 * EE, 0, 0);
                float a = 0.f;
#pragma unroll
                for (int d = 0; d < DH; d += 4) {
                    float4 k4 = *(const float4*)(kv + d);
                    a = fmaf(qh[d],     k4.x, a);
                    a = fmaf(qh[d + 1], k4.y, a);
                    a = fmaf(qh[d + 2], k4.z, a);
                    a = fmaf(qh[d + 3], k4.w, a);
                }
                s = a * scale;
            }
            s_scores[h][n] = s;
        }
    }
    __syncthreads();

    // ---- P4: per-head softmax stats (wave per head), scores -> exp ----
    {
        const int w = tid >> 5, lane = tid & 31;
        float mx = -__builtin_inff();
        for (int n = lane; n < NN; n += 32) mx = fmaxf(mx, s_scores[w][n]);
        for (int off = 16; off > 0; off >>= 1) mx = fmaxf(mx, __shfl_xor(mx, off, 32));
        float sum = 0.f;
        for (int n = lane; n < NN; n += 32) {
            float e = __expf(s_scores[w][n] - mx);
            s_scores[w][n] = e;
            sum += e;
        }
        for (int off = 16; off > 0; off >>= 1) sum += __shfl_xor(sum, off, 32);
        if (lane == 0) s_hsum[w] = sum;
    }
    __syncthreads();

    // ---- P5: ctx = attn @ v  (thread per embed col; coalesced v stream) ----
    {
        const int h = tid >> 5;
        const float* vb = vbuf + (size_t)b * NN * EE + tid;
        float a = 0.f;
        for (int n = 0; n < NN; ++n) a = fmaf(s_scores[h][n], vb[(size_t)n * EE], a);
        s_cx[tid] = a / s_hsum[h];
    }
    __syncthreads();

    // ---- P6: out proj + pointer query ----
    {
        float a = bo[tid];
        for (int i = 0; i < EE; ++i) a = fmaf(s_cx[i], Wo[i * EE + tid], a);
        s_o[tid] = a;
    }
    __syncthreads();
    {
        float a = bpq[tid];
        for (int i = 0; i < EE; ++i) a = fmaf(s_o[i], Wpq[i * EE + tid], a);
        s_qp[tid] = a;
    }
    __syncthreads();

    // ---- P7: pointer logits u = 10*tanh(qp . kp), 8 threads per node ----
    {
        const int g = tid >> 3, j = tid & 7;
        const float* qj = &s_qp[j * DH];
        for (int n = g; n < NN; n += 32) {
            const float* kp = kpbuf + ((size_t)b * NN + n) * EE + j * DH;
            __builtin_prefetch(kp + 32 * EE, 0, 0);
            float a = 0.f;
#pragma unroll
            for (int d = 0; d < DH; d += 4) {
                float4 k4 = *(const float4*)(kp + d);
                a = fmaf(qj[d],     k4.x, a);
                a = fmaf(qj[d + 1], k4.y, a);
                a = fmaf(qj[d + 2], k4.z, a);
                a = fmaf(qj[d + 3], k4.w, a);
            }
            // reduce across the 8 lanes of this group (within one wave32)
            a += __shfl_xor(a, 1, 32);
            a += __shfl_xor(a, 2, 32);
            a += __shfl_xor(a, 4, 32);
            if (j == 0)
                s_u[n] = s_mask[n] ? -__builtin_inff() : CLIPC * tanhf(a);
        }
    }
    __syncthreads();

    // ---- P8: block argmax (tie -> lowest index) + softmax over u ----
    {
        float bv = -__builtin_inff(); int bi = 0;
        for (int n = tid; n < NN; n += 256) {
            float v = s_u[n];
            if (v > bv) { bv = v; bi = n; }
        }
        s_rv[tid] = bv; s_ri[tid] = bi;
    }
    __syncthreads();
    for (int s = 128; s > 0; s >>= 1) {
        if (tid < s) {
            float ov = s_rv[tid + s]; int oi = s_ri[tid + s];
            if (ov > s_rv[tid] || (ov == s_rv[tid] && oi < s_ri[tid])) {
                s_rv[tid] = ov; s_ri[tid] = oi;
            }
        }
        __syncthreads();
    }
    const float umax = s_rv[0];
    const int   node = s_ri[0];
    __syncthreads();
    {
        float sum = 0.f;
        for (int n = tid; n < NN; n += 256) sum += __expf(s_u[n] - umax);
        s_rv[tid] = sum;
    }
    __syncthreads();
    for (int s = 128; s > 0; s >>= 1) {
        if (tid < s) s_rv[tid] += s_rv[tid + s];
        __syncthreads();
    }
    const float invsum = 1.0f / s_rv[0];
    {
        float* pout = probs_out + ((size_t)t * BB + b) * NN;
        for (int n = tid; n < NN; n += 256)
            pout[n] = __expf(s_u[n] - umax) * invsum;
    }

    // ---- P9: state update ----
    if (node != 0)
        xsum[(size_t)b * EE + tid] -= x[((size_t)b * NN + node) * EE + tid];
    if (tid == 0) {
        capacity[b] = cap - demand[(size_t)b * NN + node];
        current[b]  = node;
        route_out[(size_t)t * BB + b] = (float)node;
        if (node != 0 && !visited[(size_t)b * NN + node]) {
            visited[(size_t)b * NN + node] = 1;
            cntArr[b] -= 1;
        }
    }
}

// ---------------------------------------------------------------------------
extern "C" void kernel_launch(void* const* d_in, const int* in_sizes, int n_in,
                              void* d_out, int out_size, void* d_ws, size_t ws_size,
                              hipStream_t stream)
{
    (void)in_sizes; (void)n_in; (void)out_size; (void)ws_size;

    const float* x     = (const float*)d_in[0];
    const float* demand= (const float*)d_in[1];
    const float* cap0  = (const float*)d_in[2];
    const float* Wc    = (const float*)d_in[3];
    const float* bc    = (const float*)d_in[4];
    const float* Wqkv  = (const float*)d_in[5];
    const float* bqkv  = (const float*)d_in[6];
    const float* Wo    = (const float*)d_in[7];
    const float* bo    = (const float*)d_in[8];
    const float* Wpq   = (const float*)d_in[9];
    const float* bpq   = (const float*)d_in[10];
    const float* Wpk   = (const float*)d_in[11];
    const float* bpk   = (const float*)d_in[12];
    // d_in[13] = num_heads (=8, hardcoded)

    // Workspace layout
    const size_t ME = (size_t)MM * EE;                 // 67,108,864 elems
    float* kbuf  = (float*)d_ws;
    float* vbuf  = kbuf + ME;
    float* kpbuf = vbuf + ME;
    float* xsum  = kpbuf + ME;
    float* capacity = xsum + (size_t)BB * EE;
    int*   current  = (int*)(capacity + BB);
    int*   cntArr   = current + BB;
    unsigned char* visited = (unsigned char*)(cntArr + BB);

    // Precompute k_in / v_in / kp with WMMA GEMMs (memory-bound)
    dim3 gg(MM / 64, EE / 64);
    gemm_wmma_f32<<<gg, 128, 0, stream>>>(x, Wqkv, 3 * EE, EE,     bqkv, EE,     kbuf,  MM, EE, EE);
    gemm_wmma_f32<<<gg, 128, 0, stream>>>(x, Wqkv, 3 * EE, 2 * EE, bqkv, 2 * EE, vbuf,  MM, EE, EE);
    gemm_wmma_f32<<<gg, 128, 0, stream>>>(x, Wpk,  EE,     0,      bpk,  0,      kpbuf, MM, EE, EE);

    init_state<<<BB, 256, 0, stream>>>(x, cap0, xsum, capacity, current, cntArr, visited);

    float* probs = (float*)d_out;
    float* route = probs + (size_t)TT * BB * NN;
    for (int t = 0; t < TT; ++t) {
        decode_step<<<BB, 256, 0, stream>>>(x, demand, Wc, bc, Wqkv, bqkv, Wo, bo,
                                            Wpq, bpq, kbuf, vbuf, kpbuf,
                                            xsum, capacity, current, cntArr,
                                            visited, probs, route, t);
    }
}